// QFeatureMapOneHot_32890859552772
// MI455X (gfx1250) — compile-verified
//
#include <hip/hip_runtime.h>
#include <hip/hip_bf16.h>

typedef float v4f __attribute__((ext_vector_type(4)));

#define ROW_LEN   65536                       // 2^16 fp32 per row
#define THREADS   256                         // 8 wave32 per block
#define CHUNKS    (ROW_LEN / (THREADS * 4))   // 64 float4 chunks per thread
#define CHUNK_STRIDE_BYTES (THREADS * 16)     // 4096 B between a thread's chunks

// Emit CHUNKS async zero-stores from one 64-bit base using compile-time
// immediate offsets (IOFFSET is signed 24-bit; max here is 63*4096 = 258048).
// Template recursion guarantees the "i" constraint sees a constant expression.
template <int C>
__device__ __forceinline__ void async_zero_chunks(unsigned long long gaddr,
                                                  unsigned lds_addr)
{
    asm volatile(
        "global_store_async_from_lds_b128 %0, %1, off offset:%2 th:TH_STORE_NT"
        :
        : "v"(gaddr), "v"(lds_addr), "i"(C * CHUNK_STRIDE_BYTES)
        : "memory");
    if constexpr (C + 1 < CHUNKS) async_zero_chunks<C + 1>(gaddr, lds_addr);
}

__global__ __launch_bounds__(THREADS)
void QFeatureMapOneHot_kernel(const int* __restrict__ in, float* __restrict__ out)
{
    __shared__ float lds_zero[THREADS * 4];   // 4 KB slab of zeros
    const int tid = threadIdx.x;
    const int row = blockIdx.x;

    // Zero my private 16-byte LDS slot; __syncthreads emits s_wait_dscnt 0 +
    // barrier, so the async DMA engine is guaranteed to read zeros.
    v4f z = {0.f, 0.f, 0.f, 0.f};
    *reinterpret_cast<v4f*>(&lds_zero[tid * 4]) = z;
    __syncthreads();

    // 32-bit LDS byte address of my slot (low 32 bits of the flat pointer are
    // the LDS offset per the gfx1250 aperture mapping).
    const unsigned lds_addr =
        (unsigned)(unsigned long long)(&lds_zero[tid * 4]);

    // One-hot index for this row: addresses are wave-uniform (blockIdx only),
    // so this lowers to s_load_b512 + SALU. First digit is most significant.
    const int* __restrict__ r = in + row * 16;
    int idx = 0;
#pragma unroll
    for (int j = 0; j < 16; ++j)
        idx = (idx << 1) | (r[j] & 1);

    float* __restrict__ rowout = out + (size_t)row * ROW_LEN;

    // Bulk zero-fill: 64 back-to-back async b128 stores per thread, all from
    // one base register + immediate offsets (512 B per wave instruction,
    // tracked on ASYNCcnt, NT policy since 1 GiB >> 192 MB L2).
    const unsigned long long gaddr = (unsigned long long)(rowout + tid * 4);
    async_zero_chunks<0>(gaddr, lds_addr);

    // Drain this wave's async stores, then the thread that owns the chunk
    // containing the hot element (it issued that chunk's zero store itself)
    // overwrites a single dword with 1.0.
    asm volatile("s_wait_asynccnt 0x0" ::: "memory");
    if (((idx >> 2) & (THREADS - 1)) == tid)
        __builtin_nontemporal_store(1.0f, rowout + idx);
}

extern "C" void kernel_launch(void* const* d_in, const int* in_sizes, int n_in,
                              void* d_out, int out_size, void* d_ws, size_t ws_size,
                              hipStream_t stream)
{
    const int* in  = (const int*)d_in[0];
    float*     out = (float*)d_out;
    const int  rows = in_sizes[0] / 16;       // 4096
    QFeatureMapOneHot_kernel<<<rows, THREADS, 0, stream>>>(in, out);
}